// DampedInterpolation_32134945309332
// MI455X (gfx1250) — compile-verified
//
#include <hip/hip_runtime.h>
#include <stdint.h>

#define T48 48
#define NB 10
#define HW 16384
#define NCOL (NB*HW)           // 163840 columns
#define NGRP (NCOL/16)         // 10240 groups of 16 columns
#define CPLANE (T48*HW)        // 786432 floats per band
#define NELEM (7864320.0f)     // 10*48*16384
#define EPSF 1e-6f
#define TOLF 1e-3f
#define NBLK (NGRP/8)          // 1280 iterate blocks

typedef __attribute__((ext_vector_type(2))) float v2f;
typedef __attribute__((ext_vector_type(8))) float v8f;

// ---- workspace layout (bytes) ----
#define OFF_W      0            // 48*48 f32 = 9216
#define OFF_MASK   16384        // 16384 u64 = 131072
#define OFF_DONE   147456       // int
#define OFF_PART   151552       // 1280*2 f32 = 10240
#define OFF_F      262144       // NELEM f32 = 31457280

// ---------------------------------------------------------------------------
// W = (I + 0.1 * D^T D)^-1 via Gauss-Jordan in f64 (tridiagonal, well-cond.)
// ---------------------------------------------------------------------------
__global__ void setup_w_kernel(float* __restrict__ W, int* __restrict__ done_flag) {
  __shared__ double aug[T48][2*T48];
  __shared__ double fac[T48];
  int tid = threadIdx.x;  // 128 threads
  for (int idx = tid; idx < T48*2*T48; idx += 128) {
    int r = idx / (2*T48), c = idx % (2*T48);
    double v;
    if (c < T48) {
      if (r == c)                       v = 1.0 + 0.1 * ((r==0 || r==T48-1) ? 1.0 : 2.0);
      else if (c == r+1 || c == r-1)    v = -0.1;
      else                              v = 0.0;
    } else {
      v = ((c - T48) == r) ? 1.0 : 0.0;
    }
    aug[r][c] = v;
  }
  if (tid == 0) *done_flag = 0;
  __syncthreads();
  for (int p = 0; p < T48; ++p) {
    double piv = aug[p][p];
    __syncthreads();
    for (int c = tid; c < 2*T48; c += 128) aug[p][c] /= piv;
    __syncthreads();
    for (int r = tid; r < T48; r += 128) fac[r] = aug[r][p];
    __syncthreads();
    for (int idx = tid; idx < T48*2*T48; idx += 128) {
      int r = idx / (2*T48), c = idx % (2*T48);
      if (r != p) aug[r][c] -= fac[r] * aug[p][c];
    }
    __syncthreads();
  }
  for (int idx = tid; idx < T48*T48; idx += 128) {
    int r = idx / T48, c = idx % T48;
    W[r*T48 + c] = (float)aug[r][T48 + c];
  }
}

// ---------------------------------------------------------------------------
// init: mask bits, pixel_avg, x0 -> d_out, f = W*(m*S2) -> ws
// one thread per column (band, pixel); 48-deep series in registers
// ---------------------------------------------------------------------------
__global__ void init_kernel(const float* __restrict__ S2,
                            const int* __restrict__ cloud,
                            const float* __restrict__ Wg,
                            float* __restrict__ x,
                            float* __restrict__ f,
                            unsigned long long* __restrict__ maskbits) {
  __shared__ float Wl[T48*T48];
  int tid = threadIdx.x;  // 256
  for (int i = tid; i < T48*T48; i += 256) Wl[i] = Wg[i];
  __syncthreads();

  int col = blockIdx.x * 256 + tid;
  int c = col >> 14;
  int pixel = col & (HW - 1);
  const float* s2p = S2 + (size_t)c * CPLANE + pixel;

  float s2[T48];
  unsigned long long mask = 0ull;
  float msum = 0.f, ssum = 0.f;
#pragma unroll
  for (int t = 0; t < T48; ++t) {
    s2[t] = s2p[(size_t)t * HW];
    int isc = (cloud[(size_t)t * HW + pixel] == 1) ? 1 : 0;
    mask |= ((unsigned long long)isc) << t;
    float mf = (float)isc;
    msum += mf;
    ssum += mf * s2[t];
  }
  float avg = ssum / (msum + EPSF);

  float* xp = x + (size_t)c * CPLANE + pixel;
#pragma unroll
  for (int t = 0; t < T48; ++t) {
    bool m = (mask >> t) & 1ull;
    xp[(size_t)t * HW] = m ? s2[t] : avg;   // x0 = m*S2 + avg*(1-m)
    s2[t] = m ? s2[t] : 0.f;                // m*S2 for f
  }

  float* fp = f + (size_t)c * CPLANE + pixel;
  for (int t = 0; t < T48; ++t) {
    float acc = 0.f;
#pragma unroll
    for (int s = 0; s < T48; ++s) acc = fmaf(Wl[t*T48 + s], s2[s], acc);
    fp[(size_t)t * HW] = acc;
  }
  if (c == 0) maskbits[pixel] = mask;
}

// ---------------------------------------------------------------------------
// iterate segment: ITERS iterations of x <- f + W*(m_ .* x), WMMA f32 16x16x4
// wave handles 16 columns; x/f tiles live in VGPRs for the whole segment;
// C-layout -> B-layout relayout through a per-wave LDS tile.
// ---------------------------------------------------------------------------
template<int ITERS, bool CHECK>
__global__ void iterate_kernel(float* __restrict__ x,
                               const float* __restrict__ f,
                               const float* __restrict__ Wg,
                               const unsigned long long* __restrict__ maskbits,
                               const int* __restrict__ done_flag,
                               float* __restrict__ partials) {
  if (*done_flag) return;   // uniform
  __shared__ __align__(16) float Wl[T48*T48];
  __shared__ __align__(16) float xt[8][T48*16];
  __shared__ float red[2][256];

  int tid = threadIdx.x;  // 256 = 8 waves
  for (int i = tid; i < T48*T48; i += 256) Wl[i] = Wg[i];

  int wave = tid >> 5, lane = tid & 31;
  int n = lane & 15, h = lane >> 4;
  int grp = blockIdx.x * 8 + wave;
  int band = grp >> 10;
  int pixel = ((grp & 1023) << 4) + n;
  const size_t cb = (size_t)band * CPLANE + pixel;
  unsigned long long mask = maskbits[pixel];
  __syncthreads();  // Wl ready

  v8f fr[3], xc[3];
#pragma unroll
  for (int tt = 0; tt < 3; ++tt)
#pragma unroll
    for (int v = 0; v < 8; ++v) {
      int t = tt*16 + v + 8*h;                 // C-layout row
      fr[tt][v] = f[cb + (size_t)t * HW];
      xc[tt][v] = x[cb + (size_t)t * HW];
    }

  float* tile = &xt[wave][0];
  float sd = 0.f, sa = 0.f;

  for (int it = 0; it < ITERS; ++it) {
    // masked x (m_ .* x) into LDS tile, C-layout addressing
#pragma unroll
    for (int tt = 0; tt < 3; ++tt)
#pragma unroll
      for (int v = 0; v < 8; ++v) {
        int t = tt*16 + v + 8*h;
        float val = ((mask >> t) & 1ull) ? 0.f : xc[tt][v];
        tile[t*16 + n] = val;
      }
    __syncthreads();
    // B-layout read: lane holds X[s=4k+2h], X[s=4k+2h+1] for column n
    v2f bt[12];
#pragma unroll
    for (int k = 0; k < 12; ++k) {
      int s0 = 4*k + 2*h;
      bt[k].x = tile[s0*16 + n];
      bt[k].y = tile[(s0+1)*16 + n];
    }
    __syncthreads();
    // D = W * Xm + F, three 16x16 tiles, K=48 as 12 x WMMA(16x16x4 f32)
    v8f nx[3];
#pragma unroll
    for (int tt = 0; tt < 3; ++tt) {
      v8f acc = fr[tt];
#pragma unroll
      for (int k = 0; k < 12; ++k) {
        int wi = (tt*16 + n) * T48 + 4*k + 2*h;   // A-layout: row = lane%16
        v2f a; a.x = Wl[wi]; a.y = Wl[wi + 1];
        acc = __builtin_amdgcn_wmma_f32_16x16x4_f32(
            false, a, false, bt[k], (short)0, acc, false, false);
      }
      nx[tt] = acc;
    }
    if (CHECK && it == ITERS - 1) {
#pragma unroll
      for (int tt = 0; tt < 3; ++tt)
#pragma unroll
        for (int v = 0; v < 8; ++v) {
          sd += fabsf(nx[tt][v] - xc[tt][v]);
          sa += fabsf(nx[tt][v]);
        }
    }
#pragma unroll
    for (int tt = 0; tt < 3; ++tt) xc[tt] = nx[tt];
  }

#pragma unroll
  for (int tt = 0; tt < 3; ++tt)
#pragma unroll
    for (int v = 0; v < 8; ++v) {
      int t = tt*16 + v + 8*h;
      x[cb + (size_t)t * HW] = xc[tt][v];
    }

  if (CHECK) {
    red[0][tid] = sd; red[1][tid] = sa;
    __syncthreads();
    for (int off = 128; off > 0; off >>= 1) {
      if (tid < off) {
        red[0][tid] += red[0][tid + off];
        red[1][tid] += red[1][tid + off];
      }
      __syncthreads();
    }
    if (tid == 0) {
      partials[blockIdx.x*2 + 0] = red[0][0];
      partials[blockIdx.x*2 + 1] = red[1][0];
    }
  }
}

// ---------------------------------------------------------------------------
// convergence check: deterministic fixed-order reduction of 1280 partials
// ---------------------------------------------------------------------------
__global__ void check_kernel(const float* __restrict__ partials, int* __restrict__ done_flag) {
  if (*done_flag) return;   // uniform
  __shared__ float red[2][256];
  int tid = threadIdx.x;    // 256
  float sd = 0.f, sa = 0.f;
  for (int j = 0; j < 5; ++j) {
    int i = tid * 5 + j;    // 1280 = 256*5
    sd += partials[i*2 + 0];
    sa += partials[i*2 + 1];
  }
  red[0][tid] = sd; red[1][tid] = sa;
  __syncthreads();
  for (int off = 128; off > 0; off >>= 1) {
    if (tid < off) {
      red[0][tid] += red[0][tid + off];
      red[1][tid] += red[1][tid + off];
    }
    __syncthreads();
  }
  if (tid == 0) {
    float rel = (red[0][0] / NELEM) / ((red[1][0] / NELEM) + EPSF);
    if (rel < TOLF) *done_flag = 1;
  }
}

extern "C" void kernel_launch(void* const* d_in, const int* in_sizes, int n_in,
                              void* d_out, int out_size, void* d_ws, size_t ws_size,
                              hipStream_t stream) {
  (void)in_sizes; (void)n_in; (void)out_size; (void)ws_size;
  const float* S2   = (const float*)d_in[0];
  const int* cloud  = (const int*)d_in[1];
  char* ws = (char*)d_ws;
  float* W    = (float*)(ws + OFF_W);
  unsigned long long* maskbits = (unsigned long long*)(ws + OFF_MASK);
  int* done   = (int*)(ws + OFF_DONE);
  float* part = (float*)(ws + OFF_PART);
  float* f    = (float*)(ws + OFF_F);
  float* x    = (float*)d_out;

  setup_w_kernel<<<1, 128, 0, stream>>>(W, done);
  init_kernel<<<NCOL/256, 256, 0, stream>>>(S2, cloud, W, x, f, maskbits);

  // iter 0 + check, then 9 segments of 5 iters + check, then iters 46..49
  iterate_kernel<1, true><<<NBLK, 256, 0, stream>>>(x, f, W, maskbits, done, part);
  check_kernel<<<1, 256, 0, stream>>>(part, done);
  for (int seg = 1; seg <= 9; ++seg) {
    iterate_kernel<5, true><<<NBLK, 256, 0, stream>>>(x, f, W, maskbits, done, part);
    check_kernel<<<1, 256, 0, stream>>>(part, done);
  }
  iterate_kernel<4, false><<<NBLK, 256, 0, stream>>>(x, f, W, maskbits, done, part);
}